// FourierBlock_20675972563179
// MI455X (gfx1250) — compile-verified
//
#include <hip/hip_runtime.h>
#include <hip/hip_bf16.h>

// ---------------------------------------------------------------------------
// FourierBlock on gfx1250 (MI455X), all heavy math as V_WMMA_F32_16X16X4_F32.
//
// Pipeline:
//   per b: step1_dft64 : Y = D64 @ q_b            (real in, complex out; 2 real GEMMs)
//          step2_dft64 : X = W64 @ (twiddle * Y)  (complex GEMM, scatter f<=2048)
//   energy_kernel      : score[f] = sum_{b,h,i} |X|^2, score[0] = -inf
//   topk_kernel        : iterative argmax -> idx[64] (descending, low-index ties)
//   coef_kernel        : out_sel = x_sel @ (W_re + i W_im), packed as irfft coeffs
//   basis_kernel       : basis[t][2j]=cos, [2j+1]=sin of 2*pi*f_j*t/L (exact phase mod)
//   irfft_gemm         : out_b = basis(4096x128) @ coef_b(128x512)  (WMMA GEMM)
// ---------------------------------------------------------------------------

typedef __attribute__((ext_vector_type(2))) float v2f;
typedef __attribute__((ext_vector_type(8))) float v8f;

#define PI_D 3.14159265358979323846

__device__ __forceinline__ v8f wmma4(v2f a, v2f b, v8f c) {
  // D = A(16x4, f32) * B(4x16, f32) + C(16x16, f32)
  return __builtin_amdgcn_wmma_f32_16x16x4_f32(
      /*neg_a=*/false, a, /*neg_b=*/false, b,
      /*c_mod=*/(short)0, c, /*reuse_a=*/false, /*reuse_b=*/false);
}

// ---------------- workspace layout (in floats) ----------------
// Y (reused per batch): [re|im][64][32768]
constexpr size_t Y_RE_OFF  = 0;
constexpr size_t Y_IM_OFF  = Y_RE_OFF + (size_t)64 * 32768;          // 2,097,152
constexpr size_t X_OFF     = Y_IM_OFF + (size_t)64 * 32768;          // 4,194,304
// X: [b][f<=2048][512][re,im] = 4*2049*512*2
constexpr size_t SCORE_OFF = X_OFF + (size_t)4 * 2049 * 512 * 2;     // 12,587,008
constexpr size_t IDX_OFF   = SCORE_OFF + 2052;                       // (int) idx[64]
constexpr size_t COEF_OFF  = IDX_OFF + 64;                           // [4][128][512]
constexpr size_t BASIS_OFF = COEF_OFF + (size_t)4 * 128 * 512;       // [4096][128]

// ---------------------------------------------------------------------------
// Step 1: inner 64-pt DFT over n1 for one batch b.
//   qb viewed as Q[n1][c], c = n2*512 + h*64 + i  (contiguous slab rows).
//   Y{re,im}[k1][c] = sum_n1 e^{-2pi i k1 n1/64} * Q[n1][c]
// One wave per 16x16 tile; K = 64 -> 16 wmma per accumulator, 2 accumulators.
// ---------------------------------------------------------------------------
__global__ void __launch_bounds__(32)
step1_dft64(const float* __restrict__ qb, float* __restrict__ Yre,
            float* __restrict__ Yim) {
  const int lane = threadIdx.x;
  const int sub  = lane >> 4;        // 0: K pair {0,1}, 1: K pair {2,3}
  const int l15  = lane & 15;
  const int m0   = blockIdx.y * 16;  // k1 tile
  const int n0   = blockIdx.x * 16;  // c tile
  const int row  = m0 + l15;         // A row (k1)
  const int col  = n0 + l15;         // B col (c)

  v8f cre = {}; v8f cim = {};
  for (int kk = 0; kk < 64; kk += 4) {
    const int ka = kk + sub * 2;     // this lane's first K column
    float s0, c0, s1, c1;
    const int p0 = (row * ka) & 63;
    const int p1 = (row * (ka + 1)) & 63;
    __sincosf((float)p0 * (float)(2.0 * PI_D / 64.0), &s0, &c0);
    __sincosf((float)p1 * (float)(2.0 * PI_D / 64.0), &s1, &c1);
    v2f are = {  c0,  c1 };          // Re e^{-i t} = cos
    v2f aim = { -s0, -s1 };          // Im e^{-i t} = -sin
    v2f bv  = { qb[(size_t)ka * 32768 + col],
                qb[(size_t)(ka + 1) * 32768 + col] };
    cre = wmma4(are, bv, cre);
    cim = wmma4(aim, bv, cim);
  }
#pragma unroll
  for (int j = 0; j < 8; ++j) {
    const int r = m0 + j + sub * 8;
    Yre[(size_t)r * 32768 + col] = cre[j];
    Yim[(size_t)r * 32768 + col] = cim[j];
  }
}

// ---------------------------------------------------------------------------
// Step 2: twiddle + outer 64-pt DFT over n2, for one batch b.
//   For fixed k1: Z[n2][hi] = Y[k1][n2*512+hi] * e^{-2pi i n2 k1/4096}
//   X[64*k2 + k1][hi] = sum_n2 e^{-2pi i k2 n2/64} * Z[n2][hi]
// Complex GEMM -> 4 wmma per K-step. Store only frequencies f <= 2048.
// ---------------------------------------------------------------------------
__global__ void __launch_bounds__(32)
step2_dft64(const float* __restrict__ Yre, const float* __restrict__ Yim,
            float* __restrict__ Xout /* [2049][512][2] for this b */) {
  const int lane = threadIdx.x;
  const int sub  = lane >> 4;
  const int l15  = lane & 15;
  const int n0   = blockIdx.x * 16;  // hi tile (0..511)
  const int m0   = blockIdx.y * 16;  // k2 tile (0..63)
  const int k1   = blockIdx.z;       // 0..63
  const int row  = m0 + l15;         // k2
  const int col  = n0 + l15;         // hi

  const float* yre = Yre + (size_t)k1 * 32768;
  const float* yim = Yim + (size_t)k1 * 32768;

  v8f cre = {}; v8f cim = {};
  for (int kk = 0; kk < 64; kk += 4) {
    const int na = kk + sub * 2;     // first n2 handled by this lane
    // A = e^{-2pi i k2 n2 / 64}
    float s0, c0, s1, c1;
    const int p0 = (row * na) & 63;
    const int p1 = (row * (na + 1)) & 63;
    __sincosf((float)p0 * (float)(2.0 * PI_D / 64.0), &s0, &c0);
    __sincosf((float)p1 * (float)(2.0 * PI_D / 64.0), &s1, &c1);
    v2f are     = {  c0,  c1 };
    v2f aim     = { -s0, -s1 };
    v2f aim_neg = {  s0,  s1 };
    // B = Z rows na, na+1 at column col, with twiddle e^{-2pi i n2 k1/4096}
    float zre[2], zim[2];
#pragma unroll
    for (int d = 0; d < 2; ++d) {
      const int n2 = na + d;
      const float yr = yre[(size_t)n2 * 512 + col];
      const float yi = yim[(size_t)n2 * 512 + col];
      float ts, tc;
      const int p = (n2 * k1) & 4095;
      __sincosf((float)p * (float)(2.0 * PI_D / 4096.0), &ts, &tc);
      zre[d] = yr * tc + yi * ts;    // (yr + i yi)(tc - i ts)
      zim[d] = yi * tc - yr * ts;
    }
    v2f bre = { zre[0], zre[1] };
    v2f bim = { zim[0], zim[1] };
    cre = wmma4(are, bre, cre);
    cre = wmma4(aim_neg, bim, cre);  // - A_im * Z_im
    cim = wmma4(are, bim, cim);
    cim = wmma4(aim, bre, cim);      // + A_im * Z_re
  }
#pragma unroll
  for (int j = 0; j < 8; ++j) {
    const int k2 = m0 + j + sub * 8;
    const int f  = 64 * k2 + k1;
    if (f <= 2048) {
      float* p = Xout + ((size_t)f * 512 + col) * 2;
      p[0] = cre[j];
      p[1] = cim[j];
    }
  }
}

// ---------------------------------------------------------------------------
// Energy per frequency: score[f] = sum_{b,h,i} |X|^2 ; score[0] = -inf.
// ---------------------------------------------------------------------------
__global__ void __launch_bounds__(256)
energy_kernel(const float* __restrict__ X, float* __restrict__ score) {
  const int f = blockIdx.x;          // 0..2048
  __shared__ float red[256];
  float acc = 0.f;
  for (int t = threadIdx.x; t < 4 * 512; t += 256) {
    const int b  = t >> 9;
    const int hi = t & 511;
    const float* p = X + (((size_t)b * 2049 + f) * 512 + hi) * 2;
    acc += p[0] * p[0] + p[1] * p[1];
  }
  red[threadIdx.x] = acc;
  __syncthreads();
  for (int s = 128; s > 0; s >>= 1) {
    if ((int)threadIdx.x < s) red[threadIdx.x] += red[threadIdx.x + s];
    __syncthreads();
  }
  if (threadIdx.x == 0)
    score[f] = (f == 0) ? -INFINITY : red[0];
}

// ---------------------------------------------------------------------------
// Top-64, descending, lowest-index tie-break (matches jax.lax.top_k order;
// order matters because W is indexed by rank).
// ---------------------------------------------------------------------------
__global__ void __launch_bounds__(256)
topk_kernel(const float* __restrict__ score, int* __restrict__ idx) {
  __shared__ float sc[2049];
  __shared__ float rv[256];
  __shared__ int   ri[256];
  const int tid = threadIdx.x;
  for (int f = tid; f < 2049; f += 256) sc[f] = score[f];
  __syncthreads();
  for (int k = 0; k < 64; ++k) {
    float bv = -INFINITY;
    int   bi = 1 << 30;
    for (int f = tid; f < 2049; f += 256) {
      if (sc[f] > bv) { bv = sc[f]; bi = f; }  // ascending f -> low-index ties
    }
    rv[tid] = bv; ri[tid] = bi;
    __syncthreads();
    for (int s = 128; s > 0; s >>= 1) {
      if (tid < s) {
        if (rv[tid + s] > rv[tid] ||
            (rv[tid + s] == rv[tid] && ri[tid + s] < ri[tid])) {
          rv[tid] = rv[tid + s];
          ri[tid] = ri[tid + s];
        }
      }
      __syncthreads();
    }
    if (tid == 0) {
      idx[k] = ri[0];
      sc[ri[0]] = -INFINITY;
    }
    __syncthreads();
  }
}

// ---------------------------------------------------------------------------
// out_sel[b,j,h,o] = sum_i x_sel[b,j,h,i] * (W_re + i W_im)[j,h,i,o]
// packed directly as irfft coefficients:
//   coef[b][2j]  [h*64+o] =  s * Re(out_sel)       (cos column)
//   coef[b][2j+1][h*64+o] = -s * Im(out_sel)       (sin column; 0 at Nyquist)
//   s = 2/L (1/L at f = L/2)
// ---------------------------------------------------------------------------
__global__ void __launch_bounds__(256)
coef_kernel(const float* __restrict__ X, const float* __restrict__ Wre,
            const float* __restrict__ Wim, const int* __restrict__ idx,
            float* __restrict__ coef) {
  const int gid = blockIdx.x * 256 + threadIdx.x;  // 0..131071
  const int o = gid & 63;
  const int h = (gid >> 6) & 7;
  const int j = (gid >> 9) & 63;
  const int b = gid >> 15;
  const int f = idx[j];
  const float* xp = X + (((size_t)b * 2049 + f) * 512 + h * 64) * 2;
  const float* wr = Wre + (size_t)(j * 8 + h) * 64 * 64 + o;
  const float* wi = Wim + (size_t)(j * 8 + h) * 64 * 64 + o;
  float ore = 0.f, oim = 0.f;
  for (int i = 0; i < 64; ++i) {
    const float xr = xp[2 * i], xi = xp[2 * i + 1];
    const float wre_ = wr[(size_t)i * 64], wim_ = wi[(size_t)i * 64];
    ore += xr * wre_ - xi * wim_;
    oim += xr * wim_ + xi * wre_;
  }
  const float s = (f == 2048) ? (1.0f / 4096.0f) : (2.0f / 4096.0f);
  const float ccos = s * ore;
  const float csin = (f == 2048) ? 0.0f : (-s * oim);
  float* cb = coef + (size_t)b * 128 * 512;
  cb[((size_t)(2 * j)) * 512 + h * 64 + o]     = ccos;
  cb[((size_t)(2 * j + 1)) * 512 + h * 64 + o] = csin;
}

// ---------------------------------------------------------------------------
// basis[t][2j] = cos(2*pi*f_j*t/L), basis[t][2j+1] = sin(...).
// Exact range reduction: phase index (f*t) mod 4096 before sincos.
// ---------------------------------------------------------------------------
__global__ void __launch_bounds__(256)
basis_kernel(const int* __restrict__ idx, float* __restrict__ basis) {
  const int g = blockIdx.x * 256 + threadIdx.x;  // t*128 + r, 524288 total
  const int r = g & 127;
  const int t = g >> 7;
  const int j = r >> 1;
  const int f = idx[j];
  const int p = (f * t) & 4095;
  float s, c;
  __sincosf((float)p * (float)(2.0 * PI_D / 4096.0), &s, &c);
  basis[g] = (r & 1) ? s : c;
}

// ---------------------------------------------------------------------------
// Final sparse-irfft GEMM per batch: out_b(4096x512) = basis(4096x128) @ coef_b.
// One wave per 16x16 tile, 32 wmma_f32_16x16x4 per tile. Writes d_out directly.
// ---------------------------------------------------------------------------
__global__ void __launch_bounds__(32)
irfft_gemm(const float* __restrict__ basis, const float* __restrict__ coef,
           float* __restrict__ out) {
  const int lane = threadIdx.x;
  const int sub  = lane >> 4;
  const int l15  = lane & 15;
  const int n0   = blockIdx.x * 16;  // hi tile (0..511)
  const int m0   = blockIdx.y * 16;  // t tile  (0..4095)
  const int b    = blockIdx.z;
  const float* cf = coef + (size_t)b * 128 * 512;

  v8f acc = {};
  for (int kk = 0; kk < 128; kk += 4) {
    const int ka = kk + sub * 2;
    v2f a  = { basis[(size_t)(m0 + l15) * 128 + ka],
               basis[(size_t)(m0 + l15) * 128 + ka + 1] };
    v2f bb = { cf[(size_t)ka * 512 + n0 + l15],
               cf[(size_t)(ka + 1) * 512 + n0 + l15] };
    acc = wmma4(a, bb, acc);
  }
#pragma unroll
  for (int j = 0; j < 8; ++j) {
    const int t = m0 + j + sub * 8;
    out[(size_t)b * 2097152 + (size_t)t * 512 + n0 + l15] = acc[j];
  }
}

// ---------------------------------------------------------------------------
extern "C" void kernel_launch(void* const* d_in, const int* in_sizes, int n_in,
                              void* d_out, int out_size, void* d_ws, size_t ws_size,
                              hipStream_t stream) {
  (void)in_sizes; (void)n_in; (void)out_size; (void)ws_size;
  const float* q   = (const float*)d_in[0];  // (4,4096,8,64)
  const float* Wre = (const float*)d_in[3];  // (64,8,64,64)
  const float* Wim = (const float*)d_in[4];  // (64,8,64,64)
  float* ws = (float*)d_ws;

  float* Yre   = ws + Y_RE_OFF;
  float* Yim   = ws + Y_IM_OFF;
  float* X     = ws + X_OFF;
  float* score = ws + SCORE_OFF;
  int*   idx   = (int*)(ws + IDX_OFF);
  float* coef  = ws + COEF_OFF;
  float* basis = ws + BASIS_OFF;

  // FFT per batch (Y reused; stream serializes step1(b) -> step2(b)).
  for (int b = 0; b < 4; ++b) {
    const float* qb = q + (size_t)b * 2097152;        // 4096*8*64
    float* Xb = X + (size_t)b * 2049 * 512 * 2;
    step1_dft64<<<dim3(2048, 4, 1), 32, 0, stream>>>(qb, Yre, Yim);
    step2_dft64<<<dim3(32, 4, 64), 32, 0, stream>>>(Yre, Yim, Xb);
  }

  energy_kernel<<<2049, 256, 0, stream>>>(X, score);
  topk_kernel<<<1, 256, 0, stream>>>(score, idx);
  coef_kernel<<<512, 256, 0, stream>>>(X, Wre, Wim, idx, coef);
  basis_kernel<<<2048, 256, 0, stream>>>(idx, basis);
  irfft_gemm<<<dim3(32, 256, 4), 32, 0, stream>>>(basis, coef, (float*)d_out);
}